// Codebook_clean_7928509628503
// MI455X (gfx1250) — compile-verified
//
#include <hip/hip_runtime.h>
#include <hip/hip_bf16.h>

typedef __attribute__((ext_vector_type(2))) float v2f;
typedef __attribute__((ext_vector_type(8))) float v8f;
typedef __attribute__((ext_vector_type(4))) int v4i;
typedef __attribute__((address_space(1))) v4i v4i_glob;   // global-AS int4
typedef __attribute__((address_space(3))) v4i v4i_lds;    // LDS-AS int4

#define CDIM 256
#define KCODE 1024
#define HW 4096
#define NTOT 65536
#define PADA 260   // 260 % 64 == 4 -> conflict-free wmma operand loads
#define PADG 261   // 261 % 64 == 5 -> conflict-free column reads in gather

#if __has_builtin(__builtin_amdgcn_global_load_async_to_lds_b128) && \
    __has_builtin(__builtin_amdgcn_s_wait_asynccnt)
#define USE_ASYNC 1
#else
#define USE_ASYNC 0
#endif

// ---------------- Kernel 1: esqn[k] = -0.5 * sum_c e[k][c]^2 ----------------
__global__ __launch_bounds__(256) void vq_esq_kernel(const float* __restrict__ e,
                                                     float* __restrict__ esqn) {
    int wave = threadIdx.x >> 5;
    int lane = threadIdx.x & 31;
    int k = blockIdx.x * 8 + wave;
    const float* row = e + k * CDIM;
    float s = 0.0f;
    #pragma unroll
    for (int c = 0; c < CDIM; c += 32) {
        float v = row[c + lane];
        s += v * v;
    }
    #pragma unroll
    for (int m = 16; m >= 1; m >>= 1) s += __shfl_xor(s, m, 32);
    if (lane == 0) esqn[k] = -0.5f * s;
}

// Stage a pair of 16-col codebook tiles (32 rows x 256 f32) into LDS.
__device__ __forceinline__ void stage_b_pair(float* dstbuf, const float* e,
                                             int krow0, int tid) {
    #pragma unroll
    for (int j = 0; j < 8; ++j) {
        int chunk = j * 256 + tid;        // 0..2047 (16B chunks)
        int row = chunk >> 6;             // 0..31
        int coff = (chunk & 63) << 2;     // 0,4,...,252 (floats)
        const float* g = e + (krow0 + row) * CDIM + coff;
        float* l = dstbuf + row * PADA + coff;
#if USE_ASYNC
        __builtin_amdgcn_global_load_async_to_lds_b128(
            (v4i_glob*)g, (v4i_lds*)l, 0, 0);
#else
        *(float4*)l = *(const float4*)g;
#endif
    }
}

__device__ __forceinline__ void wait_async_staging() {
#if USE_ASYNC
    __builtin_amdgcn_s_wait_asynccnt(0);
#endif
}

// ---------------- Kernel 2: WMMA score + argmax + zp ----------------
// score[n][k] = z[n]·e[k] - 0.5*|e[k]|^2 ; argmax(score) == argmin(distance).
__global__ __launch_bounds__(256) void vq_argmin_kernel(const float* __restrict__ z,
                                                        const float* __restrict__ e,
                                                        const float* __restrict__ esqn,
                                                        int* __restrict__ idx_out,
                                                        float* __restrict__ zp_out) {
    __shared__ __align__(16) float lds_a[128 * PADA];
    __shared__ __align__(16) float lds_b[2 * 32 * PADA];  // double-buffered 32-row tiles
    __shared__ float lds_esq[KCODE];

    const int tid = threadIdx.x;
    const int n0 = blockIdx.x * 128;
    const int b = n0 >> 12;       // n0 / 4096
    const int nl0 = n0 & 4095;
    const float* zbase = z + (size_t)b * CDIM * HW + nl0;

    // prologue: async-stage codebook rows [0,32) into buffer 0
    stage_b_pair(lds_b, e, 0, tid);

    // stage esqn (4KB)
    for (int i = tid; i < KCODE; i += 256) lds_esq[i] = esqn[i];

    // stage A = z, transposed NCHW->[row][c]; global runs of 128 floats per c
    for (int i = 0; i < 128; ++i) {
        int flat = i * 256 + tid;
        int c = flat >> 7;        // 0..255
        int r = flat & 127;       // 0..127
        lds_a[r * PADA + c] = zbase[(size_t)c * HW + r];
    }
    wait_async_staging();
    __syncthreads();

    // write zp (coalesced: tid -> c)
    float* zpb = zp_out + (size_t)n0 * CDIM;
    for (int i = 0; i < 128; ++i)
        zpb[i * CDIM + tid] = lds_a[i * PADA + tid];

    const int lane = tid & 31;
    const int waveId = tid >> 5;
    const int half = lane >> 4;   // K-pair select for A/B operands
    const int lcol = lane & 15;   // A: row M; B: column N; C/D: column N
    const int m0 = waveId * 16;   // 16 rows per wave

    const float* arow = lds_a + (m0 + lcol) * PADA + 2 * half;

    float bestv[8];
    int   besti[8];
    #pragma unroll
    for (int r = 0; r < 8; ++r) { bestv[r] = -3.402823466e+38f; besti[r] = 0; }

    for (int i = 0; i < 32; ++i) {          // 32 groups of 32 codebook columns
        const int k0 = i * 32;
        if (i < 31)                          // overlap next stage with compute
            stage_b_pair(lds_b + ((i + 1) & 1) * (32 * PADA), e, k0 + 32, tid);

        const float* bb = lds_b + (i & 1) * (32 * PADA);
        const float* brow0 = bb + lcol * PADA + 2 * half;
        const float* brow1 = bb + (16 + lcol) * PADA + 2 * half;

        float c0i = lds_esq[k0 + lcol];
        float c1i = lds_esq[k0 + 16 + lcol];
        v8f acc0 = {c0i, c0i, c0i, c0i, c0i, c0i, c0i, c0i};
        v8f acc1 = {c1i, c1i, c1i, c1i, c1i, c1i, c1i, c1i};

        #pragma unroll
        for (int t = 0; t < 64; ++t) {
            v2f av = *(const v2f*)(arow + 4 * t);        // shared A operand
            v2f b0 = *(const v2f*)(brow0 + 4 * t);
            v2f b1 = *(const v2f*)(brow1 + 4 * t);
            acc0 = __builtin_amdgcn_wmma_f32_16x16x4_f32(
                false, av, false, b0, (short)0, acc0, false, false);
            acc1 = __builtin_amdgcn_wmma_f32_16x16x4_f32(
                false, av, false, b1, (short)0, acc1, false, false);
        }

        #pragma unroll
        for (int r = 0; r < 8; ++r) {       // strict '>' keeps lowest index on ties
            float v0 = acc0[r];
            bool t0 = v0 > bestv[r];
            bestv[r] = t0 ? v0 : bestv[r];
            besti[r] = t0 ? (k0 + lcol) : besti[r];
            float v1 = acc1[r];
            bool t1 = v1 > bestv[r];
            bestv[r] = t1 ? v1 : bestv[r];
            besti[r] = t1 ? (k0 + 16 + lcol) : besti[r];
        }

        wait_async_staging();   // next buffer staged before anyone reads it
        __syncthreads();
    }

    // argmax reduce across the 16 lanes of each half (xor masks 1,2,4,8)
    #pragma unroll
    for (int r = 0; r < 8; ++r) {
        float v = bestv[r];
        int ix = besti[r];
        #pragma unroll
        for (int m = 1; m <= 8; m <<= 1) {
            float v2 = __shfl_xor(v, m, 32);
            int i2 = __shfl_xor(ix, m, 32);
            if (v2 > v || (v2 == v && i2 < ix)) { v = v2; ix = i2; }
        }
        if (lcol == 0) {
            int row = m0 + half * 8 + r;   // D layout: vgpr r -> M = r + 8*half
            idx_out[n0 + row] = ix;
        }
    }
}

// ---------------- Kernel 3: z_q gather + z_out NCHW scatter ----------------
__global__ __launch_bounds__(256) void vq_gather_kernel(const float* __restrict__ z,
                                                        const float* __restrict__ e,
                                                        const int* __restrict__ idx,
                                                        float* __restrict__ zout,
                                                        float* __restrict__ zq) {
    __shared__ float lds_e[128 * PADG];
    __shared__ int lds_idx[128];

    const int tid = threadIdx.x;
    const int n0 = blockIdx.x * 128;
    if (tid < 128) lds_idx[tid] = idx[n0 + tid];
    __syncthreads();

    // stage gathered embedding rows; write z_q coalesced on the way
    for (int r = 0; r < 128; ++r) {
        float v = e[(size_t)lds_idx[r] * CDIM + tid];
        lds_e[r * PADG + tid] = v;
        zq[(size_t)(n0 + r) * CDIM + tid] = v;
    }
    __syncthreads();

    // z_out in NCHW, emulating zp + (z_q - zp) bit-exactly
    const int b = n0 >> 12;
    const int nl0 = n0 & 4095;
    const float* zb = z + (size_t)b * CDIM * HW + nl0;
    float* ob = zout + (size_t)b * CDIM * HW + nl0;
    for (int i = 0; i < 128; ++i) {
        int flat = i * 256 + tid;
        int c = flat >> 7;
        int r = flat & 127;
        float zqv = lds_e[r * PADG + c];
        float zpv = zb[(size_t)c * HW + r];
        ob[(size_t)c * HW + r] = zpv + (zqv - zpv);
    }
}

extern "C" void kernel_launch(void* const* d_in, const int* in_sizes, int n_in,
                              void* d_out, int out_size, void* d_ws, size_t ws_size,
                              hipStream_t stream) {
    const float* z = (const float*)d_in[0];             // [16,256,64,64]
    const float* e = (const float*)d_in[1];             // [1024,256]
    float* out = (float*)d_out;
    float* z_out = out;                                 // [16,256,64,64] NCHW
    float* z_q = out + (size_t)NTOT * CDIM;             // [16,64,64,256]
    float* zp  = out + (size_t)2 * NTOT * CDIM;         // [16,64,64,256]

    float* esqn = (float*)d_ws;                         // 1024 floats
    int* idxb = (int*)((char*)d_ws + KCODE * sizeof(float)); // 65536 ints

    vq_esq_kernel<<<KCODE / 8, 256, 0, stream>>>(e, esqn);
    vq_argmin_kernel<<<NTOT / 128, 256, 0, stream>>>(z, e, esqn, idxb, zp);
    vq_gather_kernel<<<NTOT / 128, 256, 0, stream>>>(z, e, idxb, z_out, z_q);
}